// EBMModel_33200097198578
// MI455X (gfx1250) — compile-verified
//
#include <hip/hip_runtime.h>

// ---------------------------------------------------------------------------
// EBM forward (binning + table lookups + sigmoid) for MI455X / gfx1250.
//
// Roofline: x stream = 64MB (~2.8us at 23.3TB/s HBM). Dominant cost is the
// 2*8*64 = 1024 LDS binary-search probes per sample -> keep edges/pair_edges/w
// resident in LDS (320KB/WGP on CDNA5 makes ~209KB/block feasible), staged via
// the CDNA5 async global->LDS DMA path (ASYNCcnt + s_wait_asynccnt).
// wave32: 256-thread block = 8 waves; 1 block/WGP (LDS bound); grid-stride so
// LDS staging amortizes over 512 samples/block.
// ---------------------------------------------------------------------------

#define N_FEAT  64
#define N_BINS  255
#define N_PAIRS 32
#define BATCH   262144

#define TPB     256
#define BLOCKS  512
#define SPB     (BATCH / BLOCKS)   // 512 samples per block
#define ITERS   (SPB / TPB)        // 2

typedef float v4f __attribute__((ext_vector_type(4)));
typedef int   v4i __attribute__((ext_vector_type(4)));

// Builtin signature (from round-1 diagnostic): param0 = v4i addrspace(1)*
// (clang prints AS1 as `__device__`), so order is (global src, lds dst, imm, imm).
typedef __attribute__((address_space(1))) v4i gas_v4i;  // global
typedef __attribute__((address_space(3))) v4i las_v4i;  // LDS

// ---- async global->LDS copy (128b), with safe fallback --------------------
__device__ __forceinline__ void g2lds_b128(const void* g, void* l) {
#if __has_builtin(__builtin_amdgcn_global_load_async_to_lds_b128)
  __builtin_amdgcn_global_load_async_to_lds_b128(
      (gas_v4i*)g, (las_v4i*)l, /*imm offset=*/0, /*imm cpol=*/0);
#else
  *(v4f*)l = *(const v4f*)g;
#endif
}

__device__ __forceinline__ void async_wait_all() {
#if __has_builtin(__builtin_amdgcn_s_wait_asynccnt)
  __builtin_amdgcn_s_wait_asynccnt(0);
#elif __has_builtin(__builtin_amdgcn_global_load_async_to_lds_b128)
  asm volatile("s_wait_asynccnt 0" ::: "memory");
#endif
}

// ---- fixed 8-step branchless upper_bound over 255 sorted floats ------------
// Returns count of E[j] <= v, in [0,255]. Probes pos+s-1 never exceed 254,
// so the unpadded 255-entry row works with exactly 8 steps.
// Two independent searches interleaved for ILP (dual dependency chains).
__device__ __forceinline__ void search2(const float* E1, const float* E2,
                                        float v, int& c1, int& c2) {
  int p1 = 0, p2 = 0;
#pragma unroll
  for (int s = 128; s > 0; s >>= 1) {
    float m1 = E1[p1 + s - 1];   // ds_load_b32
    float m2 = E2[p2 + s - 1];   // ds_load_b32
    if (m1 <= v) p1 += s;
    if (m2 <= v) p2 += s;
  }
  c1 = p1;
  c2 = p2;
}

__global__ __launch_bounds__(TPB, 1)
void ebm_logits_kernel(const float* __restrict__ x,
                       const float* __restrict__ edges,
                       const float* __restrict__ w,
                       const float* __restrict__ pair_edges,
                       const int*   __restrict__ pairs,
                       const float* __restrict__ tables,
                       const float* __restrict__ bias,
                       float* __restrict__ out) {
  // LDS: 65280 + 65280 + 65536 + 17408 = 213,504 B (< 320KB/WGP on CDNA5)
  __shared__ __attribute__((aligned(16))) float sE[N_FEAT * N_BINS];
  __shared__ __attribute__((aligned(16))) float sP[N_FEAT * N_BINS];
  __shared__ __attribute__((aligned(16))) float sW[N_FEAT * (N_BINS + 1)];
  // per-thread pair-bin scratch; stride 68B -> bank = (17*tid + f/4) % 64,
  // 17 odd => all 32 lanes hit distinct banks.
  __shared__ unsigned char sPid[TPB * 68];

  const int tid = threadIdx.x;

  // ---- stage edges / pair_edges / w into LDS via async DMA ----------------
  {
    const v4f* gE = (const v4f*)edges;
    const v4f* gP = (const v4f*)pair_edges;
    const v4f* gW = (const v4f*)w;
    v4f* lE = (v4f*)sE;
    v4f* lP = (v4f*)sP;
    v4f* lW = (v4f*)sW;
    constexpr int EV4 = (N_FEAT * N_BINS) / 4;        // 4080
    constexpr int WV4 = (N_FEAT * (N_BINS + 1)) / 4;  // 4096
    for (int i = tid; i < EV4; i += TPB) {
      g2lds_b128(gE + i, lE + i);
      g2lds_b128(gP + i, lP + i);
    }
    for (int i = tid; i < WV4; i += TPB) {
      g2lds_b128(gW + i, lW + i);
    }
  }
  async_wait_all();
  __syncthreads();

  const float b0 = bias[0];

  for (int it = 0; it < ITERS; ++it) {
    const int s = blockIdx.x * SPB + it * TPB + tid;
    const v4f* xs = (const v4f*)(x + (size_t)s * N_FEAT);

    float acc = b0;

    // 64 features: 8 independent search chains per q-iteration (4 feats x 2
    // tables) -> enough ILP to cover LDS latency at 8 waves/WGP.
#pragma unroll 1
    for (int q = 0; q < N_FEAT / 4; ++q) {
      v4f xv = __builtin_nontemporal_load(xs + q);  // stream x, don't cache
      float vv[4] = {xv.x, xv.y, xv.z, xv.w};
#pragma unroll
      for (int c = 0; c < 4; ++c) {
        const int f = q * 4 + c;
        int c1, c2;
        search2(sE + f * N_BINS, sP + f * N_BINS, vv[c], c1, c2);
        const int bin1 = (c1 < N_BINS) ? (c1 + 1) : 0;  // OOV -> 0
        const int bin2 = (c2 < N_BINS) ? (c2 + 1) : 0;
        sPid[tid * 68 + f] = (unsigned char)bin2;       // private region: no barrier
        acc += sW[(f << 8) + bin1];
      }
    }

    // pair tables: 8MB, L2-resident (192MB L2) -> random b32 gathers are fine
#pragma unroll 1
    for (int p = 0; p < N_PAIRS; ++p) {
      const int a = pairs[2 * p];       // uniform -> scalar loads
      const int b = pairs[2 * p + 1];
      const int li = sPid[tid * 68 + a];
      const int ri = sPid[tid * 68 + b];
      acc += tables[((size_t)p << 16) + (li << 8) + ri];
    }

    const float r = 1.0f / (1.0f + __expf(-acc));
    __builtin_nontemporal_store(r, out + s);
  }
}

extern "C" void kernel_launch(void* const* d_in, const int* in_sizes, int n_in,
                              void* d_out, int out_size, void* d_ws, size_t ws_size,
                              hipStream_t stream) {
  (void)in_sizes; (void)n_in; (void)out_size; (void)d_ws; (void)ws_size;
  const float* x          = (const float*)d_in[0];
  const float* edges      = (const float*)d_in[1];
  const float* w          = (const float*)d_in[2];
  const float* pair_edges = (const float*)d_in[3];
  const int*   pairs      = (const int*)d_in[4];
  const float* tables     = (const float*)d_in[5];
  const float* bias       = (const float*)d_in[6];
  float* out = (float*)d_out;

  ebm_logits_kernel<<<dim3(BLOCKS), dim3(TPB), 0, stream>>>(
      x, edges, w, pair_edges, pairs, tables, bias, out);
}